// MWNO_CZ_34746285425039
// MI455X (gfx1250) — compile-verified
//
#include <hip/hip_runtime.h>
#include <math.h>

// ---------------------------------------------------------------------------
// Multiwavelet neural operator forward pass for MI455X (gfx1250, wave32).
// Heavy math (truncated DFT, complex mode mixing, truncated iDFT) runs on
// V_WMMA_F32_16X16X4_F32 tiles; ck = C*K = 16 is the natural M dimension.
// ---------------------------------------------------------------------------

typedef float v2f __attribute__((ext_vector_type(2)));
typedef float v8f __attribute__((ext_vector_type(8)));

#define PI2F 6.283185307179586f

__device__ __forceinline__ v8f wmma4(v2f a, v2f b, v8f c) {
  // D(16x16,f32) = A(16x4,f32) * B(4x16,f32) + C
  return __builtin_amdgcn_wmma_f32_16x16x4_f32(
      /*neg_a=*/false, a, /*neg_b=*/false, b,
      /*c_mod=*/(short)0, c, /*reuse_a=*/false, /*reuse_b=*/false);
}

// ---------------------------------------------------------------------------
// Layout transforms: (B,N,C,K) <-> planar (B, ck, N), ch = c*K + k.
// ---------------------------------------------------------------------------
__global__ void k_tin(const float* __restrict__ x, float* __restrict__ xp, int n) {
  int idx = blockIdx.x * blockDim.x + threadIdx.x;
  int tot = 32 * 16 * n;
  if (idx >= tot) return;
  int pos = idx % n;
  int ch  = (idx / n) & 15;
  int b   = idx / (n * 16);
  xp[idx] = x[((size_t)b * n + pos) * 16 + ch];
}

__global__ void k_tout(const float* __restrict__ xp, float* __restrict__ out, int n) {
  int idx = blockIdx.x * blockDim.x + threadIdx.x;
  int tot = 32 * 16 * n;
  if (idx >= tot) return;
  int ch  = idx & 15;
  int pos = (idx >> 4) % n;
  int b   = idx / (n * 16);
  out[idx] = xp[((size_t)b * 16 + ch) * n + pos];
}

// ---------------------------------------------------------------------------
// Decomposition: xa = [x_even, x_odd]; d = xa@ec_d, s = xa@ec_s  (8->4 per c)
// planar in (len 2h) -> planar d,s (len h)
// ---------------------------------------------------------------------------
__global__ void k_decomp(const float* __restrict__ xin,
                         const float* __restrict__ ec_d,
                         const float* __restrict__ ec_s,
                         float* __restrict__ dout, float* __restrict__ sout,
                         int h) {
  int idx = blockIdx.x * blockDim.x + threadIdx.x;
  int tot = 32 * 16 * h;
  if (idx >= tot) return;
  int j  = idx % h;
  int co = (idx / h) & 15;
  int b  = idx / (h * 16);
  int c = co >> 2, kp = co & 3;
  int n = h * 2;
  const float* base = xin + ((size_t)(b * 16 + c * 4)) * n + 2 * j;
  float d = 0.f, s = 0.f;
#pragma unroll
  for (int kk = 0; kk < 4; ++kk) {
    v2f eo = *(const v2f*)(base + (size_t)kk * n);  // {x_even, x_odd}
    d += ec_d[kk * 4 + kp] * eo.x + ec_d[(4 + kk) * 4 + kp] * eo.y;
    s += ec_s[kk * 4 + kp] * eo.x + ec_s[(4 + kk) * 4 + kp] * eo.y;
  }
  dout[idx] = d;
  sout[idx] = s;
}

// ---------------------------------------------------------------------------
// Truncated forward rDFT via WMMA: xf[b,ch,f] = sum_t sig[b,ch,t] e^{-2πi f t/h}
// Split-K over `nc` t-chunks; partial 16x16 complex tiles to workspace.
// grid.x = B * nft * 2(sig) * nc, block = 32 (one wave).
// h and nc are powers of two, so for h>=4 every chunk is fully in bounds ->
// straight-line global_load_b64 path; guards exist only for h in {1,2}.
// ---------------------------------------------------------------------------
__global__ void k_fwd_dft(const float* __restrict__ dsig, const float* __restrict__ ssig,
                          float* __restrict__ xdre, float* __restrict__ xdim,
                          float* __restrict__ xsre, float* __restrict__ xsim,
                          int h, int l, int nft, int nc) {
  int bid = blockIdx.x;
  int chunk = bid % nc; bid /= nc;
  int sig = bid & 1;   bid >>= 1;
  int ft = bid % nft;
  int b  = bid / nft;

  const float* src = sig ? ssig : dsig;
  float* ore = sig ? xsre : xdre;
  float* oim = sig ? xsim : xdim;

  int lane = threadIdx.x;
  int hi = lane >> 4, lo = lane & 15;
  int f = ft * 16 + lo;          // B-fragment column (frequency)
  const float* arow = src + ((size_t)(b * 16 + lo)) * h;  // A row = channel lo
  const float* ap = arow + 2 * hi;                        // lane's K base

  int tlen = h / nc;             // h, nc powers of two
  int t0 = chunk * tlen;
  int tend = t0 + tlen;
  int mask = h - 1;
  float w = PI2F / (float)h;

  v8f accRe = {0, 0, 0, 0, 0, 0, 0, 0};
  v8f accIm = {0, 0, 0, 0, 0, 0, 0, 0};

  if (h >= 16) {
    // tlen is a multiple of 16 for all these levels: 4 K-steps per group,
    // one prefetch per group, loads pipelined.
    for (int t = t0; t < tend; t += 16) {
      __builtin_prefetch(ap + t + 128, 0, 3);
#pragma unroll
      for (int u = 0; u < 16; u += 4) {
        int tt = t + u;
        v2f a = *(const v2f*)(ap + tt);
        int ta = tt + 2 * hi;
        int r0 = (f * ta) & mask;
        int r1 = (f * (ta + 1)) & mask;
        float s0, c0, s1, c1;
        __sincosf((float)r0 * w, &s0, &c0);
        __sincosf((float)r1 * w, &s1, &c1);
        v2f bc; bc.x = c0;  bc.y = c1;
        v2f bs; bs.x = -s0; bs.y = -s1;   // rfft: Im = -sum x sin
        accRe = wmma4(a, bc, accRe);
        accIm = wmma4(a, bs, accIm);
      }
    }
  } else if (h >= 4) {
    for (int t = t0; t < tend; t += 4) {
      v2f a = *(const v2f*)(ap + t);
      int ta = t + 2 * hi;
      int r0 = (f * ta) & mask;
      int r1 = (f * (ta + 1)) & mask;
      float s0, c0, s1, c1;
      __sincosf((float)r0 * w, &s0, &c0);
      __sincosf((float)r1 * w, &s1, &c1);
      v2f bc; bc.x = c0;  bc.y = c1;
      v2f bs; bs.x = -s0; bs.y = -s1;
      accRe = wmma4(a, bc, accRe);
      accIm = wmma4(a, bs, accIm);
    }
  } else {
    // h in {1,2}: single masked K-step
    int ta = 2 * hi;
    v2f a;
    a.x = (ta     < h) ? arow[ta]     : 0.f;
    a.y = (ta + 1 < h) ? arow[ta + 1] : 0.f;
    int r0 = (f * ta) & mask;
    int r1 = (f * (ta + 1)) & mask;
    float s0, c0, s1, c1;
    __sincosf((float)r0 * w, &s0, &c0);
    __sincosf((float)r1 * w, &s1, &c1);
    v2f bc; bc.x = c0;  bc.y = c1;
    v2f bs; bs.x = -s0; bs.y = -s1;
    accRe = wmma4(a, bc, accRe);
    accIm = wmma4(a, bs, accIm);
  }

  if (f < l) {
    size_t obase = (((size_t)chunk * 32 + b) * 16) * 64 + f;
#pragma unroll
    for (int r = 0; r < 8; ++r) {
      int M = r + 8 * hi;        // channel row of D tile
      ore[obase + (size_t)M * 64] = accRe[r];
      oim[obase + (size_t)M * 64] = accIm[r];
    }
  }
}

// ---------------------------------------------------------------------------
// Per-mode complex mixing:  Ud = Xd@wA + Xs@wB ;  Us = Xd@wC
// A = (16 batch rows) x (16 ch), B = w[i][o][f].  Chunk partials reduced on load.
// grid.x = l * 2(batch tiles), block = 32.
// ---------------------------------------------------------------------------
__global__ void k_mix(const float* __restrict__ xdre, const float* __restrict__ xdim,
                      const float* __restrict__ xsre, const float* __restrict__ xsim,
                      const float* __restrict__ wAr, const float* __restrict__ wAi,
                      const float* __restrict__ wBr, const float* __restrict__ wBi,
                      const float* __restrict__ wCr, const float* __restrict__ wCi,
                      float* __restrict__ udr, float* __restrict__ udi,
                      float* __restrict__ usr, float* __restrict__ usi,
                      int nc) {
  int f  = blockIdx.x >> 1;
  int bt = blockIdx.x & 1;
  int lane = threadIdx.x;
  int hi = lane >> 4, lo = lane & 15;

  v8f PudR = {0,0,0,0,0,0,0,0}, PudI = {0,0,0,0,0,0,0,0};
  v8f PusR = {0,0,0,0,0,0,0,0}, PusI = {0,0,0,0,0,0,0,0};

#pragma unroll
  for (int kk = 0; kk < 4; ++kk) {
    int i0 = 4 * kk + 2 * hi;    // channel (K) slots for this lane
    int i1 = i0 + 1;
    int b = bt * 16 + lo;        // A row = batch
    v2f dre = {0, 0}, dim = {0, 0}, sre = {0, 0}, sim = {0, 0};
    for (int ch = 0; ch < nc; ++ch) {
      size_t p0 = (((size_t)ch * 32 + b) * 16 + i0) * 64 + f;
      size_t p1 = (((size_t)ch * 32 + b) * 16 + i1) * 64 + f;
      dre.x += xdre[p0]; dre.y += xdre[p1];
      dim.x += xdim[p0]; dim.y += xdim[p1];
      sre.x += xsre[p0]; sre.y += xsre[p1];
      sim.x += xsim[p0]; sim.y += xsim[p1];
    }
    size_t q0 = ((size_t)i0 * 16 + lo) * 64 + f;   // w[i][o][f], o = lo
    size_t q1 = ((size_t)i1 * 16 + lo) * 64 + f;
    v2f ar = {wAr[q0], wAr[q1]}, ai = {wAi[q0], wAi[q1]};
    v2f br = {wBr[q0], wBr[q1]}, bi = {wBi[q0], wBi[q1]};
    v2f cr = {wCr[q0], wCr[q1]}, ci = {wCi[q0], wCi[q1]};
    v2f nai = {-ai.x, -ai.y}, nbi = {-bi.x, -bi.y}, nci = {-ci.x, -ci.y};

    PudR = wmma4(dre, ar, PudR);  PudR = wmma4(dim, nai, PudR);
    PudR = wmma4(sre, br, PudR);  PudR = wmma4(sim, nbi, PudR);
    PudI = wmma4(dre, ai, PudI);  PudI = wmma4(dim, ar, PudI);
    PudI = wmma4(sre, bi, PudI);  PudI = wmma4(sim, br, PudI);
    PusR = wmma4(dre, cr, PusR);  PusR = wmma4(dim, nci, PusR);
    PusI = wmma4(dre, ci, PusI);  PusI = wmma4(dim, cr, PusI);
  }

#pragma unroll
  for (int r = 0; r < 8; ++r) {
    int M = r + 8 * hi;
    int b = bt * 16 + M;
    size_t p = ((size_t)b * 16 + lo) * 64 + f;   // out channel o = lo
    udr[p] = PudR[r]; udi[p] = PudI[r];
    usr[p] = PusR[r]; usi[p] = PusI[r];
  }
}

// ---------------------------------------------------------------------------
// Truncated irfft via WMMA:
//   y[t] = (1/h) * sum_{f<l} wf(f) * (Xre cos(2πft/h) - Xim sin(2πft/h))
//   wf = 1 for f==0 or f==h/2 (Nyquist), else 2.
// Full K-steps (f+3 < l) are unguarded b64 loads; Nyquist can only appear in
// the tail step (l = h/2+1 levels), handled separately.
// grid.x = B * ntt * 2(sig), block = 32.
// ---------------------------------------------------------------------------
__global__ void k_idft(const float* __restrict__ udr, const float* __restrict__ udi,
                       const float* __restrict__ usr, const float* __restrict__ usi,
                       float* __restrict__ udo, float* __restrict__ uso,
                       int h, int l, int ntt) {
  int bid = blockIdx.x;
  int sig = bid & 1; bid >>= 1;
  int tt = bid % ntt;
  int b  = bid / ntt;

  const float* srcR = sig ? usr : udr;
  const float* srcI = sig ? usi : udi;
  float* dst = sig ? uso : udo;

  int lane = threadIdx.x;
  int hi = lane >> 4, lo = lane & 15;
  int t = tt * 16 + lo;
  int mask = h - 1;
  float w = PI2F / (float)h;

  v8f acc = {0, 0, 0, 0, 0, 0, 0, 0};
  size_t pbase = ((size_t)b * 16 + lo) * 64;   // A row = channel lo

  int kfull = l >> 2;                          // unguarded K-steps
  for (int kk = 0; kk < kfull; ++kk) {
    int f0 = 4 * kk + 2 * hi;
    v2f are = *(const v2f*)(srcR + pbase + f0);
    v2f aim = *(const v2f*)(srcI + pbase + f0);
    float w0 = (f0 == 0) ? 1.f : 2.f;          // Nyquist never in full steps
    are.x *= w0; are.y *= 2.f;
    aim.x *= w0; aim.y *= 2.f;
    int r0 = (f0 * t) & mask;
    int r1 = ((f0 + 1) * t) & mask;
    float s0, c0, s1, c1;
    __sincosf((float)r0 * w, &s0, &c0);
    __sincosf((float)r1 * w, &s1, &c1);
    v2f bc; bc.x = c0;  bc.y = c1;
    v2f bs; bs.x = -s0; bs.y = -s1;
    acc = wmma4(are, bc, acc);
    acc = wmma4(aim, bs, acc);
  }

  if ((kfull << 2) < l) {                      // tail K-step (l % 4 != 0)
    int f0 = 4 * kfull + 2 * hi;
    int f1 = f0 + 1;
    float w0 = (f0 == 0 || 2 * f0 == h) ? 1.f : 2.f;
    float w1 = (2 * f1 == h) ? 1.f : 2.f;
    v2f are, aim;
    are.x = (f0 < l) ? srcR[pbase + f0] * w0 : 0.f;
    are.y = (f1 < l) ? srcR[pbase + f1] * w1 : 0.f;
    aim.x = (f0 < l) ? srcI[pbase + f0] * w0 : 0.f;
    aim.y = (f1 < l) ? srcI[pbase + f1] * w1 : 0.f;
    int r0 = (f0 * t) & mask;
    int r1 = (f1 * t) & mask;
    float s0, c0, s1, c1;
    __sincosf((float)r0 * w, &s0, &c0);
    __sincosf((float)r1 * w, &s1, &c1);
    v2f bc; bc.x = c0;  bc.y = c1;
    v2f bs; bs.x = -s0; bs.y = -s1;
    acc = wmma4(are, bc, acc);
    acc = wmma4(aim, bs, acc);
  }

  if (t < h) {
    float sc = 1.f / (float)h;
#pragma unroll
    for (int r = 0; r < 8; ++r) {
      int M = r + 8 * hi;
      dst[((size_t)b * 16 + M) * h + t] = acc[r] * sc;
    }
  }
}

// ---------------------------------------------------------------------------
// Coarsest level: y = x @ t0_w.T + t0_b  (per c, over last dim K)
// ---------------------------------------------------------------------------
__global__ void k_t0(const float* __restrict__ sin_, const float* __restrict__ t0w,
                     const float* __restrict__ t0b, float* __restrict__ xout) {
  int idx = blockIdx.x * blockDim.x + threadIdx.x;
  if (idx >= 32 * 16) return;
  int co = idx & 15;
  int b  = idx >> 4;
  int c = co >> 2, kp = co & 3;
  float acc = t0b[kp];
#pragma unroll
  for (int k = 0; k < 4; ++k)
    acc += sin_[b * 16 + c * 4 + k] * t0w[kp * 4 + k];
  xout[idx] = acc;
}

// ---------------------------------------------------------------------------
// Reconstruction: v = x + Us; xc = [v, Ud]; x_e = xc@rc_e, x_o = xc@rc_o;
// interleave even/odd -> length 2h (b64 store per thread).
// ---------------------------------------------------------------------------
__global__ void k_recon(const float* __restrict__ xin, const float* __restrict__ Ud,
                        const float* __restrict__ Us, const float* __restrict__ rce,
                        const float* __restrict__ rco, float* __restrict__ xout,
                        int h) {
  int idx = blockIdx.x * blockDim.x + threadIdx.x;
  int tot = 32 * 16 * h;
  if (idx >= tot) return;
  int j  = idx % h;
  int co = (idx / h) & 15;
  int b  = idx / (h * 16);
  int c = co >> 2, kp = co & 3;
  size_t base = ((size_t)(b * 16 + c * 4)) * h + j;
  float xe = 0.f, xo = 0.f;
#pragma unroll
  for (int kk = 0; kk < 4; ++kk) {
    float v = xin[base + (size_t)kk * h] + Us[base + (size_t)kk * h];
    float u = Ud[base + (size_t)kk * h];
    xe += rce[kk * 4 + kp] * v + rce[(4 + kk) * 4 + kp] * u;
    xo += rco[kk * 4 + kp] * v + rco[(4 + kk) * 4 + kp] * u;
  }
  v2f r; r.x = xe; r.y = xo;
  *(v2f*)(xout + ((size_t)(b * 16 + co)) * (2 * h) + 2 * j) = r;
}

// ---------------------------------------------------------------------------
// Host orchestration
// ---------------------------------------------------------------------------
extern "C" void kernel_launch(void* const* d_in, const int* in_sizes, int n_in,
                              void* d_out, int out_size, void* d_ws, size_t ws_size,
                              hipStream_t stream) {
  const float* x    = (const float*)d_in[0];
  const float* ec_s = (const float*)d_in[1];
  const float* ec_d = (const float*)d_in[2];
  const float* rc_e = (const float*)d_in[3];
  const float* rc_o = (const float*)d_in[4];
  const float* t0_w = (const float*)d_in[5];
  const float* t0_b = (const float*)d_in[6];
  const float* wAr  = (const float*)d_in[7];
  const float* wAi  = (const float*)d_in[8];
  const float* wBr  = (const float*)d_in[9];
  const float* wBi  = (const float*)d_in[10];
  const float* wCr  = (const float*)d_in[11];
  const float* wCi  = (const float*)d_in[12];

  const int B = 32, N = 16384, CK = 16, NS = 14, AL = 64;

  float* ws = (float*)d_ws;
  size_t off = 0;
  float* xp   = ws + off; off += (size_t)B * CK * N;        // planar x / final recon
  float* dtmp = ws + off; off += (size_t)B * CK * (N / 2);  // detail coeffs (per level)
  float* p0   = ws + off; off += (size_t)B * CK * (N / 2);  // ping
  float* p1   = ws + off; off += (size_t)B * CK * (N / 2);  // pong
  float* Ud   = ws + off; off += (size_t)B * CK * (N - 1);  // stacked per-level Ud
  float* Us   = ws + off; off += (size_t)B * CK * (N - 1);  // stacked per-level Us
  const size_t xfsz = (size_t)16 * B * CK * AL;             // 16 chunk partials max
  float* xdre = ws + off; off += xfsz;
  float* xdim = ws + off; off += xfsz;
  float* xsre = ws + off; off += xfsz;
  float* xsim = ws + off; off += xfsz;
  const size_t usz = (size_t)B * CK * AL;
  float* udr = ws + off; off += usz;
  float* udi = ws + off; off += usz;
  float* usr = ws + off; off += usz;
  float* usi = ws + off; off += usz;
  (void)ws_size; (void)in_sizes; (void)n_in; (void)out_size;

  { // (B,N,C,K) -> planar
    int tot = B * CK * N;
    k_tin<<<(tot + 255) / 256, 256, 0, stream>>>(x, xp, N);
  }

  // ---- decomposition sweep ----
  for (int i = 0; i < NS; ++i) {
    int n = N >> i, h = n >> 1;
    int l = (AL < h / 2 + 1) ? AL : (h / 2 + 1);
    const float* xin = (i == 0) ? xp : ((i & 1) ? p0 : p1);
    float* sout = (i & 1) ? p1 : p0;

    int tot = B * CK * h;
    k_decomp<<<(tot + 255) / 256, 256, 0, stream>>>(xin, ec_d, ec_s, dtmp, sout, h);

    int nc = (h >= 8192) ? 16 : (h >= 4096) ? 8 : (h >= 2048) ? 4 : (h >= 1024) ? 2 : 1;
    int nft = (l + 15) / 16;
    k_fwd_dft<<<B * nft * 2 * nc, 32, 0, stream>>>(dtmp, sout, xdre, xdim, xsre, xsim,
                                                   h, l, nft, nc);
    k_mix<<<l * 2, 32, 0, stream>>>(xdre, xdim, xsre, xsim,
                                    wAr, wAi, wBr, wBi, wCr, wCi,
                                    udr, udi, usr, usi, nc);
    int ntt = (h + 15) / 16;
    size_t lb = (size_t)B * CK * (size_t)(N - (N >> i));
    k_idft<<<B * ntt * 2, 32, 0, stream>>>(udr, udi, usr, usi,
                                           Ud + lb, Us + lb, h, l, ntt);
  }

  // ---- coarsest linear map (s_13 lives in p1, seed recon in p0) ----
  k_t0<<<2, 256, 0, stream>>>(p1, t0_w, t0_b, p0);

  // ---- reconstruction sweep ----
  for (int i = NS - 1; i >= 0; --i) {
    int h = N >> (i + 1);
    const float* xin = ((i + 1) & 1) ? p1 : p0;
    float* xout = (i == 0) ? xp : ((i & 1) ? p1 : p0);
    size_t lb = (size_t)B * CK * (size_t)(N - (N >> i));
    int tot = B * CK * h;
    k_recon<<<(tot + 255) / 256, 256, 0, stream>>>(xin, Ud + lb, Us + lb,
                                                   rc_e, rc_o, xout, h);
  }

  { // planar -> (B,N,C,K)
    int tot = B * CK * N;
    k_tout<<<(tot + 255) / 256, 256, 0, stream>>>(xp, (float*)d_out, N);
  }
}